// FlexDecodingModule_11003706212958
// MI455X (gfx1250) — compile-verified
//
#include <hip/hip_runtime.h>
#include <hip/hip_bf16.h>

// FlexDecoding on MI455X (gfx1250):
// Decode-step attention is rewritten to avoid materializing the KV cache:
//   scores[b,h,s] = tokens_ext[b,s,:] . qk[b,h,:],  qk = Wk_h^T q  (pre-scaled)
//   out[b,h,:]    = Wv_h @ (sum_s probs[b,h,s] * tokens_ext[b,s,:])
// => two streaming f32-WMMA passes over tokens (134 MB each) + one pass over
// each weight matrix. ~340 MB total HBM traffic, ~2.3 GFLOP, exact f32 math.

#define B_ 4
#define S_ 4096
#define DIM_ 2048
#define H_ 16
#define HD_ 128
#define SP1_ 4097  // S+1

typedef __attribute__((ext_vector_type(2))) float v2f;
typedef __attribute__((ext_vector_type(8))) float v8f;

__device__ __forceinline__ v8f wmma_f32_16x16x4(v2f a, v2f b, v8f c) {
    // 8-arg pattern: (neg_a, A, neg_b, B, c_mod, C, reuse_a, reuse_b)
    return __builtin_amdgcn_wmma_f32_16x16x4_f32(
        false, a, false, b, (short)0, c, false, false);
}

// ---------------------------------------------------------------------------
// y[b][o] = dot(W[o, :], x[b, (head_mode ? (o>>7)*DIM slice of per-head x : x)])
// One wave per output row o; lanes stride the 2048-long dot; all 4 batches
// accumulated per weight read (W is read exactly once from HBM).
// grid = 256 blocks x 256 threads (8 waves/block, 2048 rows total).
__global__ void k_gemv_wt(const float* __restrict__ W, const float* __restrict__ x,
                          float* __restrict__ y, int head_mode) {
    int wave = threadIdx.x >> 5;
    int lane = threadIdx.x & 31;
    int o = blockIdx.x * 8 + wave;                 // 0..2047
    const float* wrow = W + (size_t)o * DIM_;
    int xoff = head_mode ? ((o >> 7) * DIM_) : 0;  // per-head slice of t_bar
    int xstr = head_mode ? (H_ * DIM_) : DIM_;
    float a0 = 0.f, a1 = 0.f, a2 = 0.f, a3 = 0.f;
    for (int e = lane; e < DIM_; e += 32) {
        float w = wrow[e];
        a0 += w * x[0 * xstr + xoff + e];
        a1 += w * x[1 * xstr + xoff + e];
        a2 += w * x[2 * xstr + xoff + e];
        a3 += w * x[3 * xstr + xoff + e];
    }
    #pragma unroll
    for (int m = 16; m >= 1; m >>= 1) {
        a0 += __shfl_xor(a0, m); a1 += __shfl_xor(a1, m);
        a2 += __shfl_xor(a2, m); a3 += __shfl_xor(a3, m);
    }
    if (lane == 0) {
        y[0 * DIM_ + o] = a0; y[1 * DIM_ + o] = a1;
        y[2 * DIM_ + o] = a2; y[3 * DIM_ + o] = a3;
    }
}

// ---------------------------------------------------------------------------
// qk[b][h][e] = scale * sum_d q[b, h*HD+d] * Wk[h*HD+d, e]
// Thread per (h,e): coalesced over e; q value is wave-uniform (scalar load).
// grid = 128 x 256 (32768 threads).
__global__ void k_qk(const float* __restrict__ Wk, const float* __restrict__ q,
                     float* __restrict__ qkout) {
    int t = blockIdx.x * 256 + threadIdx.x;
    int h = t >> 11;
    int e = t & (DIM_ - 1);
    float a0 = 0.f, a1 = 0.f, a2 = 0.f, a3 = 0.f;
    for (int d = 0; d < HD_; d++) {
        float w = Wk[((size_t)(h * HD_ + d)) * DIM_ + e];
        int qi = h * HD_ + d;
        a0 += w * q[0 * DIM_ + qi];
        a1 += w * q[1 * DIM_ + qi];
        a2 += w * q[2 * DIM_ + qi];
        a3 += w * q[3 * DIM_ + qi];
    }
    const float scale = 0.08838834764831845f;  // 1/sqrt(128)
    qkout[((size_t)(0 * H_ + h)) * DIM_ + e] = a0 * scale;
    qkout[((size_t)(1 * H_ + h)) * DIM_ + e] = a1 * scale;
    qkout[((size_t)(2 * H_ + h)) * DIM_ + e] = a2 * scale;
    qkout[((size_t)(3 * H_ + h)) * DIM_ + e] = a3 * scale;
}

// ---------------------------------------------------------------------------
// scores[b,h,s] for s in [0,S): per-(b, 16-row tile) wave does
//   D(16x16) += A(tokens 16x4) x B(qk^T 4x16) over K = 2048.
// A layout (f32 16x4): lane L holds (M=L&15, K=2*(L>>4)+{0,1}) -> float2.
// B layout (f32 4x16): lane L holds (K=2*(L>>4)+{0,1}, N=L&15) -> float2.
// C: element (M = v + 8*(L>>4), N = L&15); M->s row, N->head.
// grid = B*256 blocks x 32 threads.
__global__ void k_scores_wmma(const float* __restrict__ tokens,
                              const float* __restrict__ qk,
                              float* __restrict__ scores) {
    int blk = blockIdx.x;
    int b = blk >> 8;
    int tile = blk & 255;
    int lane = threadIdx.x;
    int r = lane & 15;
    int sel = lane >> 4;
    const float* At = tokens + ((size_t)(b * S_ + tile * 16 + r)) * DIM_ + 2 * sel;
    const float* Bq = qk + ((size_t)(b * H_ + r)) * DIM_ + 2 * sel;
    v8f c = {};
    for (int k = 0; k < DIM_; k += 4) {
        v2f a = *(const v2f*)(At + k);   // 8B aligned: even float offset
        v2f bb = *(const v2f*)(Bq + k);
        c = wmma_f32_16x16x4(a, bb, c);
    }
    // lane's column N = r is the head; rows M = v + 8*sel are s offsets
    size_t base = ((size_t)(b * H_ + r)) * SP1_ + (size_t)tile * 16 + 8 * sel;
    #pragma unroll
    for (int v = 0; v < 8; v++) scores[base + v] = c[v];
}

// scores[b,h,S] = qk[b,h,:] . token[b,:]  (qk already pre-scaled)
// grid = 64 x 32.
__global__ void k_scores_last(const float* __restrict__ qk,
                              const float* __restrict__ token,
                              float* __restrict__ scores) {
    int bh = blockIdx.x;       // b*16 + h
    int b = bh >> 4;
    int lane = threadIdx.x;
    const float* qr = qk + (size_t)bh * DIM_;
    const float* tr = token + (size_t)b * DIM_;
    float acc = 0.f;
    for (int e = lane; e < DIM_; e += 32) acc += qr[e] * tr[e];
    #pragma unroll
    for (int m = 16; m >= 1; m >>= 1) acc += __shfl_xor(acc, m);
    if (lane == 0) scores[(size_t)bh * SP1_ + S_] = acc;
}

// ---------------------------------------------------------------------------
// In-place softmax over the 4097 scores of each (b,h). grid = 64 x 256.
__global__ void k_softmax(float* __restrict__ p) {
    int bh = blockIdx.x;
    float* row = p + (size_t)bh * SP1_;
    int tid = threadIdx.x;
    int lane = tid & 31, wave = tid >> 5;
    __shared__ float red[8];
    float m = -3.402823466e38f;
    for (int i = tid; i < SP1_; i += 256) m = fmaxf(m, row[i]);
    #pragma unroll
    for (int s = 16; s >= 1; s >>= 1) m = fmaxf(m, __shfl_xor(m, s));
    if (lane == 0) red[wave] = m;
    __syncthreads();
    float mx = red[0];
    #pragma unroll
    for (int w = 1; w < 8; w++) mx = fmaxf(mx, red[w]);
    __syncthreads();
    float sum = 0.f;
    for (int i = tid; i < SP1_; i += 256) {
        float e = expf(row[i] - mx);
        row[i] = e;
        sum += e;
    }
    #pragma unroll
    for (int s = 16; s >= 1; s >>= 1) sum += __shfl_xor(sum, s);
    if (lane == 0) red[wave] = sum;
    __syncthreads();
    float tot = 0.f;
    #pragma unroll
    for (int w = 0; w < 8; w++) tot += red[w];
    float inv = 1.f / tot;
    for (int i = tid; i < SP1_; i += 256) row[i] *= inv;
}

// ---------------------------------------------------------------------------
// t_bar[b,h,e] = sum_{s=0..S} probs[b,h,s] * tokens_ext[b,s,e]
// WMMA: A = probs (M=16 heads x K=4 positions), B = tokens (K=4 x N=16 cols).
// A loads are scalar (4097 stride -> only 4B aligned). Tail row s=S folded in
// after the K loop via the C-fragment mapping (M = v+8*sel -> head,
// N = lane&15 -> embedding col). grid = B*128 blocks x 32 threads.
__global__ void k_tbar_wmma(const float* __restrict__ tokens,
                            const float* __restrict__ token,
                            const float* __restrict__ probs,
                            float* __restrict__ tbar) {
    int blk = blockIdx.x;
    int b = blk >> 7;
    int e0 = (blk & 127) * 16;
    int lane = threadIdx.x;
    int n = lane & 15;
    int sel = lane >> 4;
    const float* Ap = probs + ((size_t)(b * H_ + n)) * SP1_ + 2 * sel;
    const float* Bt = tokens + ((size_t)(b * S_ + 2 * sel)) * DIM_ + e0 + n;
    v8f c = {};
    for (int s = 0; s < S_; s += 4) {
        v2f a;
        a.x = Ap[s];
        a.y = Ap[s + 1];
        v2f bb;
        bb.x = Bt[(size_t)s * DIM_];
        bb.y = Bt[(size_t)s * DIM_ + DIM_];
        c = wmma_f32_16x16x4(a, bb, c);
    }
    float tk = token[(size_t)b * DIM_ + e0 + n];  // tokens_ext row s = S
    #pragma unroll
    for (int v = 0; v < 8; v++) {
        int h = v + 8 * sel;
        float pS = probs[((size_t)(b * H_ + h)) * SP1_ + S_];
        float val = c[v] + pS * tk;
        tbar[((size_t)(b * H_ + h)) * DIM_ + e0 + n] = val;
    }
}

// ---------------------------------------------------------------------------
extern "C" void kernel_launch(void* const* d_in, const int* in_sizes, int n_in,
                              void* d_out, int out_size, void* d_ws, size_t ws_size,
                              hipStream_t stream) {
    const float* tokens = (const float*)d_in[0];
    const float* token  = (const float*)d_in[1];
    const float* Wq     = (const float*)d_in[2];
    const float* Wk     = (const float*)d_in[3];
    const float* Wv     = (const float*)d_in[4];
    const float* Wo     = (const float*)d_in[5];
    float* out = (float*)d_out;
    float* ws  = (float*)d_ws;

    // Workspace layout (floats): q[8192] qk[131072] probs[262208]
    //                            tbar[131072] attn[8192]  (~2.2 MB total)
    float* q_ws     = ws;
    float* qk_ws    = ws + 8192;
    float* probs_ws = ws + 8192 + 131072;              // 139264
    float* tbar_ws  = ws + 139264 + 262208;            // 401472
    float* attn_ws  = ws + 401472 + 131072;            // 532544

    // 1) q[b] = token[b] @ Wq.T                  (Wq read once)
    k_gemv_wt<<<256, 256, 0, stream>>>(Wq, token, q_ws, 0);
    // 2) qk[b,h,:] = scale * Wk_h^T q[b,h]       (Wk read once)
    k_qk<<<128, 256, 0, stream>>>(Wk, q_ws, qk_ws);
    // 3) scores[b,h,0..S) via f32 WMMA           (tokens pass #1, 134 MB)
    k_scores_wmma<<<B_ * 256, 32, 0, stream>>>(tokens, qk_ws, probs_ws);
    // 4) scores[b,h,S] for the new token
    k_scores_last<<<B_ * H_, 32, 0, stream>>>(qk_ws, token, probs_ws);
    // 5) softmax over 4097 positions, in place
    k_softmax<<<B_ * H_, 256, 0, stream>>>(probs_ws);
    // 6) t_bar = probs @ tokens_ext via f32 WMMA (tokens pass #2, 134 MB)
    k_tbar_wmma<<<B_ * 128, 32, 0, stream>>>(tokens, token, probs_ws, tbar_ws);
    // 7) attn[b, h*HD+d] = Wv_h @ t_bar[b,h]     (Wv read once)
    k_gemv_wt<<<256, 256, 0, stream>>>(Wv, tbar_ws, attn_ws, 1);
    // 8) out = attn @ Wo.T                       (Wo read once)
    k_gemv_wt<<<256, 256, 0, stream>>>(Wo, attn_ws, out, 0);
}